// MaskConv2d_35845797053219
// MI455X (gfx1250) — compile-verified
//
#include <hip/hip_runtime.h>

typedef float v2f __attribute__((ext_vector_type(2)));
typedef float v8f __attribute__((ext_vector_type(8)));

#define B_    8
#define CIN   64
#define COUT  64
#define H_    256
#define W_    256
#define TILEW 64          // output positions (w) per block
#define TILEH 2           // output rows (h) per block
#define CICH  16          // input channels staged per K-chunk
#define NK    9           // 3x3 taps
#define NCOL  66          // staged input cols: w0-1 .. w0+64
#define NROW  4           // staged input rows: h0-1 .. h0+2
#define CIP   20          // padded per-(row,col) ci stride (16 used) -> conflict-free b64

__global__ __launch_bounds__(256)
void maskconv2d_wmma_f32(const float* __restrict__ x,
                         const int*   __restrict__ mask,
                         const float* __restrict__ wgt,
                         const float* __restrict__ bias,
                         float* __restrict__ out)
{
    // sW[((khw*4 + kk)*64 + co)*4 + dk] : A fragments; lane-half reads dk={2hi,2hi+1} as one b64
    // sX[(row*NCOL + col)*CIP + pos(ci)]: B fragments; ci group stored [0,2,1,3] so
    //                                     lane-half reads one aligned b64 at kk*4+2hi
    __shared__ float sW[NK * CICH * COUT];          // 9216 floats = 36.9 KB
    __shared__ float sX[NROW * NCOL * CIP];         // 5280 floats = 21.1 KB
    __shared__ float sB[COUT];

    const int tid  = threadIdx.x;
    const int wave = tid >> 5;
    const int lane = tid & 31;
    const int hi   = lane >> 4;
    const int ln   = lane & 15;

    const int w0 = blockIdx.x * TILEW;
    const int h0 = blockIdx.y * TILEH;
    const int b  = blockIdx.z;

    if (tid < COUT) sB[tid] = bias[tid];

    const int co_tile = (wave >> 1) * 16;   // 4 co-tiles of 16
    const int hrow    = wave & 1;           // which output row this wave computes

    v8f acc[4];
    #pragma unroll
    for (int t = 0; t < 4; ++t) acc[t] = (v8f){0.f,0.f,0.f,0.f,0.f,0.f,0.f,0.f};

    for (int cc = 0; cc < CIN / CICH; ++cc) {
        const int ci0 = cc * CICH;

        // ---- stage weights via async copy (memory -> LDS, no VGPR round-trip):
        //      wgt[co][ci][kh][kw] -> sW[((khw*4+kk)*64+co)*4 + dk]
        for (int idx = tid; idx < NK * CICH * COUT; idx += 256) {
            int dk  = idx & 3;
            int co  = (idx >> 2) & 63;
            int kq  = idx >> 8;            // khw*4 + kk
            int khw = kq >> 2;
            int kk  = kq & 3;
            const float* gp = &wgt[co * (CIN * NK) + (ci0 + kk * 4 + dk) * NK + khw];
            unsigned     lo = (unsigned)(uintptr_t)&sW[idx];   // LDS byte address = addr[31:0]
            asm volatile("global_load_async_to_lds_b32 %0, %1, off"
                         :: "v"(lo), "v"(gp) : "memory");
        }

        // ---- stage input halo tile via async copy; OOB halo slots zero-filled via DS
        //      (disjoint slots -> no DS/ASYNC ordering hazard)
        for (int idx = tid; idx < CICH * NROW * NCOL; idx += 256) {
            int col = idx % NCOL;
            int rc  = idx / NCOL;
            int row = rc & 3;
            int ci  = rc >> 2;
            int gh  = h0 - 1 + row;
            int gw  = w0 - 1 + col;
            int c    = ci & 3;
            int cpos = (ci & ~3) + ((c & 1) << 1) + (c >> 1);   // [0,2,1,3] order
            float*   lp = &sX[(row * NCOL + col) * CIP + cpos];
            if ((unsigned)gh < H_ && (unsigned)gw < W_) {
                const float* gp = &x[(((b * CIN) + (ci0 + ci)) * H_ + gh) * W_ + gw];
                unsigned     lo = (unsigned)(uintptr_t)lp;
                asm volatile("global_load_async_to_lds_b32 %0, %1, off"
                             :: "v"(lo), "v"(gp) : "memory");
            } else {
                *lp = 0.f;
            }
        }

        asm volatile("s_wait_asynccnt 0" ::: "memory");
        __syncthreads();

        // ---- compute: D[16co x 16pos] += A[16co x 4k] * B[4k x 16pos], 4 pos-tiles/wave
        for (int kh = 0; kh < 3; ++kh) {
            const int irow = hrow + kh;            // input row in sX for this wave
            for (int kw = 0; kw < 3; ++kw) {
                const int khw = kh * 3 + kw;
                #pragma unroll
                for (int kk = 0; kk < CICH / 4; ++kk) {
                    // A fragment: one aligned b64, dk = {2hi, 2hi+1}
                    v2f a = *(const v2f*)&sW[(khw * 4 + kk) * (COUT * 4)
                                             + (co_tile + ln) * 4 + 2 * hi];
                    #pragma unroll
                    for (int t = 0; t < 4; ++t) {
                        int col = t * 16 + ln + kw;          // input col = pos + kw
                        // B fragment: one aligned b64, slots {2hi, 2hi+1} = ci {hi, hi+2}
                        v2f bf = *(const v2f*)&sX[(irow * NCOL + col) * CIP
                                                  + kk * 4 + 2 * hi];
                        acc[t] = __builtin_amdgcn_wmma_f32_16x16x4_f32(
                            false, a, false, bf, (short)0, acc[t], false, false);
                    }
                }
            }
        }
        __syncthreads();
    }

    // ---- epilogue: +bias, *mask, coalesced stores along w
    const int hout = h0 + hrow;
    const int mrow = (b * H_ + hout) * W_;
    #pragma unroll
    for (int t = 0; t < 4; ++t) {
        #pragma unroll
        for (int r = 0; r < 8; ++r) {
            int   co = co_tile + r + 8 * hi;      // D layout: M = r + 8*hi
            float bi = sB[co];
            int   w  = w0 + t * 16 + ln;          // N = ln
            float m  = (float)mask[mrow + w];
            out[(((b * COUT) + co) * H_ + hout) * W_ + w] = (acc[t][r] + bi) * m;
        }
    }
}

extern "C" void kernel_launch(void* const* d_in, const int* in_sizes, int n_in,
                              void* d_out, int out_size, void* d_ws, size_t ws_size,
                              hipStream_t stream) {
    (void)in_sizes; (void)n_in; (void)d_ws; (void)ws_size; (void)out_size;
    const float* x    = (const float*)d_in[0];
    const int*   mask = (const int*)  d_in[1];
    const float* wgt  = (const float*)d_in[2];
    const float* bias = (const float*)d_in[3];
    float*       out  = (float*)d_out;

    dim3 grid(W_ / TILEW, H_ / TILEH, B_);   // 4 x 128 x 8 = 4096 blocks
    dim3 block(256);
    maskconv2d_wmma_f32<<<grid, block, 0, stream>>>(x, mask, wgt, bias, out);
}